// ScatterModule_35287451304683
// MI455X (gfx1250) — compile-verified
//
#include <hip/hip_runtime.h>

typedef __attribute__((ext_vector_type(2))) float v2f;
typedef __attribute__((ext_vector_type(8))) float v8f;

#define D_DIM 512
#define ROWS_PER_WAVE 64
#define WAVES_PER_BLOCK 8
#define ROWS_PER_BLOCK (ROWS_PER_WAVE * WAVES_PER_BLOCK) // 512

// Pass 1: zero the seg_sums scratch (aliased onto d_out) and the total[D] accumulator.
__global__ void seg_zero_kernel(float4* __restrict__ out, float4* __restrict__ total, long n4) {
    long g = (long)blockIdx.x * blockDim.x + threadIdx.x;
    if (g < n4)       out[g]   = make_float4(0.f, 0.f, 0.f, 0.f);
    if (g < D_DIM/4)  total[g] = make_float4(0.f, 0.f, 0.f, 0.f);
}

// Pass 2: scatter-add source rows into seg_sums[indices[row], :].
// One thread owns one float4 of one source row; destination row is contiguous,
// so the 4 global_atomic_add_f32 per thread coalesce across the wave.
__global__ void seg_scatter_kernel(const float4* __restrict__ src,
                                   const int* __restrict__ idx,
                                   float* __restrict__ segsum, long n4) {
    long g = (long)blockIdx.x * blockDim.x + threadIdx.x;
    if (g >= n4) return;
    int row = (int)(g >> 7);       // D/4 = 128 float4 per row
    int q   = (int)(g & 127);
    float4 v = src[g];
    int s = idx[row];
    float* dst = segsum + (size_t)s * D_DIM + (size_t)q * 4;
    atomicAdd(dst + 0, v.x);
    atomicAdd(dst + 1, v.y);
    atomicAdd(dst + 2, v.z);
    atomicAdd(dst + 3, v.w);
}

// Pass 3: total[d] += sum_s segsum[s][d]^2 via V_WMMA_F32_16X16X4_F32 with A=ones.
// D[m,n] = sum_k 1 * B[k,n]  -> every output row holds the column sums, so the
// matrix pipe performs the 4-rows-at-a-time reduction for free (exact f32).
// blockIdx.x selects a 16-column tile, blockIdx.y a 512-row slab; each of the
// 8 waves owns a 64-row sub-slab. EXEC is uniformly all-ones at every WMMA.
__global__ void seg_reduce_wmma_kernel(const float* __restrict__ segsum,
                                       float* __restrict__ total) {
    int wave = threadIdx.x >> 5;
    int lane = threadIdx.x & 31;
    int n    = lane & 15;          // output column within the 16-wide tile
    int kh   = lane >> 4;          // K half: lanes 16-31 carry K=2,3 rows
    int colBase = blockIdx.x * 16;
    long rowBase = (long)blockIdx.y * ROWS_PER_BLOCK + (long)wave * ROWS_PER_WAVE;

    v2f ones; ones.x = 1.0f; ones.y = 1.0f;   // A = 16x4 all-ones
    v8f c = {};

    const float* base = segsum + (size_t)colBase + n;
    #pragma unroll 4
    for (int s = 0; s < ROWS_PER_WAVE; s += 4) {
        long r0 = rowBase + s + 2 * kh;       // B vgpr0: K=0 (lanes 0-15) / K=2 (16-31)
        float x0 = base[(size_t)r0 * D_DIM];
        float x1 = base[(size_t)(r0 + 1) * D_DIM]; // B vgpr1: K=1 / K=3
        v2f b; b.x = x0 * x0; b.y = x1 * x1;
        // 8 args: (neg_a, A, neg_b, B, c_mod, C, reuse_a, reuse_b)
        c = __builtin_amdgcn_wmma_f32_16x16x4_f32(
                false, ones, false, b, (short)0, c, false, false);
    }
    // C vgpr0: lanes 0-15 = row M=0 (cols 0-15); rows are all identical column
    // sums, so lanes 0-15 commit one partial each.
    if (lane < 16) atomicAdd(&total[colBase + n], c[0]);
}

// Pass 4: broadcast total[D] into every output row (float4 streaming stores).
__global__ void seg_bcast_kernel(float4* __restrict__ out,
                                 const float4* __restrict__ total, long n4) {
    __shared__ float4 t[D_DIM / 4];
    if (threadIdx.x < D_DIM / 4) t[threadIdx.x] = total[threadIdx.x];
    __syncthreads();
    long g = (long)blockIdx.x * blockDim.x + threadIdx.x;
    if (g < n4) out[g] = t[g & (D_DIM / 4 - 1)];
}

extern "C" void kernel_launch(void* const* d_in, const int* in_sizes, int n_in,
                              void* d_out, int out_size, void* d_ws, size_t ws_size,
                              hipStream_t stream) {
    (void)n_in; (void)out_size; (void)ws_size;
    const float* src = (const float*)d_in[0];
    const int*   idx = (const int*)d_in[1];
    float*       out = (float*)d_out;      // doubles as seg_sums scratch
    float*     total = (float*)d_ws;       // D floats

    long N  = in_sizes[1];                 // number of rows / segments
    long n4 = (N * (long)D_DIM) / 4;       // float4 count over [N, D]

    const int threads = 256;
    unsigned blocks = (unsigned)((n4 + threads - 1) / threads);

    seg_zero_kernel<<<dim3(blocks), threads, 0, stream>>>(
        (float4*)out, (float4*)total, n4);

    seg_scatter_kernel<<<dim3(blocks), threads, 0, stream>>>(
        (const float4*)src, idx, out, n4);

    dim3 rgrid(D_DIM / 16, (unsigned)(N / ROWS_PER_BLOCK));
    seg_reduce_wmma_kernel<<<rgrid, WAVES_PER_BLOCK * 32, 0, stream>>>(out, total);

    seg_bcast_kernel<<<dim3(blocks), threads, 0, stream>>>(
        (float4*)out, (const float4*)total, n4);
}